// EGCL_84052509983239
// MI455X (gfx1250) — compile-verified
//
#include <hip/hip_runtime.h>

// ---------------------------------------------------------------------------
// EGNN EGCL layer for MI455X (gfx1250, wave32, WMMA).
// All GEMMs run on V_WMMA_F32_16X16X4_F32 (full fp32 fidelity vs reference).
// Edge kernel uses 64-edge tiles: each eW2 B-fragment feeds 4 WMMAs.
// ---------------------------------------------------------------------------

typedef float v2f __attribute__((ext_vector_type(2)));
typedef float v8f __attribute__((ext_vector_type(8)));

static constexpr int FDIM = 128;   // node feature dim
static constexpr int HDIM = 128;   // hidden dim
static constexpr int EADIM = 8;    // edge-attr dim
static constexpr int ETILE = 64;   // edges per block in edge kernel

static __device__ __forceinline__ float silu_f(float x) {
    return x / (1.0f + __expf(-x));
}
static __device__ __forceinline__ float sigmoid_f(float x) {
    return 1.0f / (1.0f + __expf(-x));
}

// D = A(16x4 f32) * B(4x16 f32) + C(16x16 f32)
static __device__ __forceinline__ v8f wmma4(v2f a, v2f b, v8f c) {
    return __builtin_amdgcn_wmma_f32_16x16x4_f32(
        /*neg_a=*/false, a, /*neg_b=*/false, b,
        /*c_mod=*/(short)0, c, /*reuse_a=*/false, /*reuse_b=*/false);
}

// ---------------------------------------------------------------------------
// Kernel 0: zero the accumulation scratch (agg[N,128] | aggT[N,3] | cnt[N])
// ---------------------------------------------------------------------------
__global__ void zero_kernel(float* __restrict__ p, size_t n) {
    size_t i = (size_t)blockIdx.x * blockDim.x + threadIdx.x;
    if (i < n) p[i] = 0.0f;
}

// ---------------------------------------------------------------------------
// Kernel 1: node precompute  Pa = h @ eW1[0:128,:],  Pb = h @ eW1[128:256,:]
// Block = 256 threads = 8 waves; block handles 16 node rows; wave w owns
// output column tile w (16 cols). Both GEMMs share the A fragment.
// ---------------------------------------------------------------------------
__global__ __launch_bounds__(256) void node_pre_kernel(
    const float* __restrict__ h, const float* __restrict__ eW1,
    float* __restrict__ Pa, float* __restrict__ Pb)
{
    const int wave = threadIdx.x >> 5;
    const int lane = threadIdx.x & 31;
    const int half = lane >> 4;
    const int l16  = lane & 15;
    const int row0 = blockIdx.x * 16;
    const int col  = wave * 16 + l16;

    const float* hrow = h + (size_t)(row0 + l16) * FDIM;
    v8f ca = {};
    v8f cb = {};
    for (int k0 = 0; k0 < FDIM; k0 += 4) {
        const int k = k0 + 2 * half;
        v2f a = *(const v2f*)(hrow + k);                       // A[M=l16][k,k+1]
        v2f b1, b2;
        b1.x = eW1[(size_t)(k    ) * HDIM + col];
        b1.y = eW1[(size_t)(k + 1) * HDIM + col];
        b2.x = eW1[(size_t)(FDIM + k    ) * HDIM + col];
        b2.y = eW1[(size_t)(FDIM + k + 1) * HDIM + col];
        ca = wmma4(a, b1, ca);
        cb = wmma4(a, b2, cb);
    }
    for (int i = 0; i < 8; ++i) {
        const int r = row0 + i + 8 * half;
        Pa[(size_t)r * HDIM + col] = ca[i];
        Pb[(size_t)r * HDIM + col] = cb[i];
    }
}

// ---------------------------------------------------------------------------
// Kernel 2: edge model. Block = 256 threads = 8 waves, 64 edges per block.
//  phase 1: per-edge gathers (row/col ids, pos_diff, dist^2, edge_attr)
//  phase 2: m1 = silu(Pa[row]+Pb[col]+d2*w_d+ea@W_e+eb1)   (float4 VALU ->LDS)
//  phase 3: m2 = silu(m1 @ eW2 + eb2)   (WMMA: 4 edge-row tiles share each B)
//  phase 4: LDS-atomic dot reductions: gate = m2@aW, tr = m2@pW2
//  phase 5: m = m2*sigmoid(gate+ab); write m; atomic scatter agg/aggT/cnt
// ---------------------------------------------------------------------------
__global__ __launch_bounds__(256) void edge_kernel(
    const float* __restrict__ pos, const int* __restrict__ ei,
    const float* __restrict__ eattr,
    const float* __restrict__ eW1, const float* __restrict__ eb1,
    const float* __restrict__ eW2, const float* __restrict__ eb2,
    const float* __restrict__ aW,  const float* __restrict__ ab,
    const float* __restrict__ pW2,
    const float* __restrict__ Pa,  const float* __restrict__ Pb,
    float* __restrict__ agg, float* __restrict__ aggT, float* __restrict__ cnt,
    float* __restrict__ m_out, int nEdges)
{
    __shared__ float m1s[ETILE][132];       // padded stride (128 + 4 dwords)
    __shared__ float m2s[ETILE][132];
    __shared__ int   rS[ETILE], cS[ETILE];
    __shared__ float pdS[ETILE][3], d2S[ETILE], eaS[ETILE][EADIM];
    __shared__ float gateAcc[ETILE], transAcc[ETILE];

    const int t  = threadIdx.x;
    const int e0 = blockIdx.x * ETILE;

    // ---- phase 1: per-edge setup ----
    if (t < ETILE) {
        const int e = e0 + t;
        const int r = ei[e];
        const int c = ei[nEdges + e];
        rS[t] = r; cS[t] = c;
        float d2 = 0.0f;
        for (int k = 0; k < 3; ++k) {
            const float pd = pos[(size_t)r * 3 + k] - pos[(size_t)c * 3 + k];
            pdS[t][k] = pd;
            d2 += pd * pd;
        }
        d2S[t] = d2;
        for (int a = 0; a < EADIM; ++a) eaS[t][a] = eattr[(size_t)e * EADIM + a];
        gateAcc[t] = 0.0f;
        transAcc[t] = 0.0f;
    }
    __syncthreads();

    const int le = t >> 2;          // edge within tile (4 threads / edge)
    const int q  = t & 3;           // 32-channel chunk index
    const int c0 = q * 32;

    // ---- phase 2: layer-1 elementwise (refactored GEMM), float4 ----
    {
        const float* pa = Pa + (size_t)rS[le] * HDIM + c0;
        const float* pb = Pb + (size_t)cS[le] * HDIM + c0;
        const float* wd = eW1 + (size_t)(2 * FDIM) * HDIM + c0;
        const float* b1 = eb1 + c0;
        const float  d2 = d2S[le];
        float ea[EADIM];
        for (int a = 0; a < EADIM; ++a) ea[a] = eaS[le][a];
        for (int j = 0; j < 8; ++j) {       // 8 float4 = 32 channels
            const int ch = j * 4;
            float4 v  = *(const float4*)(pa + ch);
            float4 vb = *(const float4*)(pb + ch);
            float4 vw = *(const float4*)(wd + ch);
            float4 ve = *(const float4*)(b1 + ch);
            float px = v.x + vb.x + ve.x + d2 * vw.x;
            float py = v.y + vb.y + ve.y + d2 * vw.y;
            float pz = v.z + vb.z + ve.z + d2 * vw.z;
            float pw = v.w + vb.w + ve.w + d2 * vw.w;
            for (int a = 0; a < EADIM; ++a) {
                float4 we = *(const float4*)(eW1 + (size_t)(2 * FDIM + 1 + a) * HDIM + c0 + ch);
                px += ea[a] * we.x;  py += ea[a] * we.y;
                pz += ea[a] * we.z;  pw += ea[a] * we.w;
            }
            float4 o;
            o.x = silu_f(px); o.y = silu_f(py); o.z = silu_f(pz); o.w = silu_f(pw);
            *(float4*)(&m1s[le][c0 + ch]) = o;
        }
    }
    __syncthreads();

    // ---- phase 3: m2 = silu(m1 @ eW2 + eb2) via WMMA (4 row tiles / wave) ----
    {
        const int wave = t >> 5;
        const int lane = t & 31;
        const int half = lane >> 4;
        const int l16  = lane & 15;
        const int col  = wave * 16 + l16;
        v8f acc0 = {}, acc1 = {}, acc2 = {}, acc3 = {};
        for (int k0 = 0; k0 < HDIM; k0 += 4) {
            const int k = k0 + 2 * half;
            v2f b; b.x = eW2[(size_t)k * HDIM + col];
                   b.y = eW2[(size_t)(k + 1) * HDIM + col];
            v2f a0 = *(const v2f*)(&m1s[ 0 + l16][k]);
            v2f a1 = *(const v2f*)(&m1s[16 + l16][k]);
            v2f a2 = *(const v2f*)(&m1s[32 + l16][k]);
            v2f a3 = *(const v2f*)(&m1s[48 + l16][k]);
            acc0 = wmma4(a0, b, acc0);
            acc1 = wmma4(a1, b, acc1);
            acc2 = wmma4(a2, b, acc2);
            acc3 = wmma4(a3, b, acc3);
        }
        const float bias = eb2[col];
        for (int i = 0; i < 8; ++i) {
            const int em = i + 8 * half;        // edge row within 16-row tile
            m2s[ 0 + em][col] = silu_f(acc0[i] + bias);
            m2s[16 + em][col] = silu_f(acc1[i] + bias);
            m2s[32 + em][col] = silu_f(acc2[i] + bias);
            m2s[48 + em][col] = silu_f(acc3[i] + bias);
        }
    }
    __syncthreads();

    // ---- phase 4: gate / translation dot products (float4 partials) ----
    {
        float gp = 0.0f, tp = 0.0f;
        for (int j = 0; j < 8; ++j) {
            const int ch = c0 + j * 4;
            float4 v  = *(const float4*)(&m2s[le][ch]);
            float4 wa = *(const float4*)(aW  + ch);
            float4 wp = *(const float4*)(pW2 + ch);
            gp += v.x * wa.x + v.y * wa.y + v.z * wa.z + v.w * wa.w;
            tp += v.x * wp.x + v.y * wp.y + v.z * wp.z + v.w * wp.w;
        }
        atomicAdd(&gateAcc[le], gp);     // ds_add_f32
        atomicAdd(&transAcc[le], tp);
    }
    __syncthreads();

    // ---- phase 5: gate, emit m, scatter segment sums ----
    {
        const float g = sigmoid_f(gateAcc[le] + ab[0]);
        const int r = rS[le];
        float* aggRow = agg + (size_t)r * HDIM;
        float* mRow   = m_out + (size_t)(e0 + le) * HDIM;
        for (int j = 0; j < 8; ++j) {
            const int ch = c0 + j * 4;
            float4 v = *(const float4*)(&m2s[le][ch]);
            float4 mv; mv.x = v.x * g; mv.y = v.y * g; mv.z = v.z * g; mv.w = v.w * g;
            *(float4*)(mRow + ch) = mv;                  // coalesced b128 store
            atomicAdd(aggRow + ch + 0, mv.x);            // global_atomic_add_f32
            atomicAdd(aggRow + ch + 1, mv.y);
            atomicAdd(aggRow + ch + 2, mv.z);
            atomicAdd(aggRow + ch + 3, mv.w);
        }
        if (q == 0) {
            const float s = g * transAcc[le];            // (m @ pW2) scalar
            atomicAdd(&aggT[(size_t)r * 3 + 0], pdS[le][0] * s);
            atomicAdd(&aggT[(size_t)r * 3 + 1], pdS[le][1] * s);
            atomicAdd(&aggT[(size_t)r * 3 + 2], pdS[le][2] * s);
            atomicAdd(&cnt[r], 1.0f);
        }
    }
}

// ---------------------------------------------------------------------------
// Kernel 3: node model + pos update.
//  o = silu(h@nW1[0:128] + agg@nW1[128:256] + nb1)   (two WMMA chains, one C)
//  h_new = o @ nW2 + nb2 + h                          (WMMA from LDS)
//  pos_new = pos + aggT / max(cnt,1)
// ---------------------------------------------------------------------------
__global__ __launch_bounds__(256) void node_out_kernel(
    const float* __restrict__ h, const float* __restrict__ pos,
    const float* __restrict__ nW1, const float* __restrict__ nb1,
    const float* __restrict__ nW2, const float* __restrict__ nb2,
    const float* __restrict__ agg, const float* __restrict__ aggT,
    const float* __restrict__ cnt,
    float* __restrict__ hOut, float* __restrict__ posOut)
{
    __shared__ float oS[16][132];
    const int t    = threadIdx.x;
    const int wave = t >> 5;
    const int lane = t & 31;
    const int half = lane >> 4;
    const int l16  = lane & 15;
    const int row0 = blockIdx.x * 16;
    const int col  = wave * 16 + l16;

    {   // layer 1
        v8f acc = {};
        const float* hrow = h   + (size_t)(row0 + l16) * FDIM;
        const float* arow = agg + (size_t)(row0 + l16) * HDIM;
        for (int k0 = 0; k0 < FDIM; k0 += 4) {
            const int k = k0 + 2 * half;
            v2f a1 = *(const v2f*)(hrow + k);
            v2f a2 = *(const v2f*)(arow + k);
            v2f b1; b1.x = nW1[(size_t)k * HDIM + col];
                    b1.y = nW1[(size_t)(k + 1) * HDIM + col];
            v2f b2; b2.x = nW1[(size_t)(FDIM + k) * HDIM + col];
                    b2.y = nW1[(size_t)(FDIM + k + 1) * HDIM + col];
            acc = wmma4(a1, b1, acc);
            acc = wmma4(a2, b2, acc);
        }
        const float bias = nb1[col];
        for (int i = 0; i < 8; ++i)
            oS[i + 8 * half][col] = silu_f(acc[i] + bias);
    }
    __syncthreads();

    {   // layer 2 + residual
        v8f acc = {};
        for (int k0 = 0; k0 < HDIM; k0 += 4) {
            const int k = k0 + 2 * half;
            v2f a = *(const v2f*)(&oS[l16][k]);
            v2f b; b.x = nW2[(size_t)k * FDIM + col];
                   b.y = nW2[(size_t)(k + 1) * FDIM + col];
            acc = wmma4(a, b, acc);
        }
        const float bias = nb2[col];
        for (int i = 0; i < 8; ++i) {
            const int node = row0 + i + 8 * half;
            hOut[(size_t)node * FDIM + col] =
                acc[i] + bias + h[(size_t)node * FDIM + col];
        }
    }

    if (t < 48) {   // pos update (unsorted segment mean)
        const int node = row0 + t / 3;
        const int d    = t % 3;
        const float cv = cnt[node];
        posOut[(size_t)node * 3 + d] =
            pos[(size_t)node * 3 + d] +
            aggT[(size_t)node * 3 + d] / fmaxf(cv, 1.0f);
    }
}

// ---------------------------------------------------------------------------
extern "C" void kernel_launch(void* const* d_in, const int* in_sizes, int n_in,
                              void* d_out, int out_size, void* d_ws, size_t ws_size,
                              hipStream_t stream)
{
    (void)n_in; (void)out_size; (void)ws_size;
    const float* h     = (const float*)d_in[0];
    const float* pos   = (const float*)d_in[1];
    const int*   ei    = (const int*)  d_in[2];
    const float* eattr = (const float*)d_in[3];
    const float* eW1   = (const float*)d_in[4];
    const float* eb1   = (const float*)d_in[5];
    const float* eW2   = (const float*)d_in[6];
    const float* eb2   = (const float*)d_in[7];
    const float* aW    = (const float*)d_in[8];
    const float* ab    = (const float*)d_in[9];
    const float* nW1   = (const float*)d_in[10];
    const float* nb1   = (const float*)d_in[11];
    const float* nW2   = (const float*)d_in[12];
    const float* nb2   = (const float*)d_in[13];
    const float* pW2   = (const float*)d_in[14];

    const int nNodes = in_sizes[0] / FDIM;   // 50000 (multiple of 16)
    const int nEdges = in_sizes[2] / 2;      // 640000 (multiple of 64)

    // workspace: Pa | Pb | agg | aggT | cnt   (all fp32)
    float* Pa   = (float*)d_ws;
    float* Pb   = Pa   + (size_t)nNodes * HDIM;
    float* agg  = Pb   + (size_t)nNodes * HDIM;
    float* aggT = agg  + (size_t)nNodes * HDIM;
    float* cnt  = aggT + (size_t)nNodes * 3;
    const size_t zeroN = (size_t)nNodes * (HDIM + 4);   // agg + aggT + cnt

    // outputs: h_new [N,128] | pos_new [N,3] | m [E,128]
    float* hOut   = (float*)d_out;
    float* posOut = hOut   + (size_t)nNodes * FDIM;
    float* mOut   = posOut + (size_t)nNodes * 3;

    zero_kernel<<<(unsigned)((zeroN + 255) / 256), 256, 0, stream>>>(agg, zeroN);
    node_pre_kernel<<<nNodes / 16, 256, 0, stream>>>(h, eW1, Pa, Pb);
    edge_kernel<<<nEdges / ETILE, 256, 0, stream>>>(pos, ei, eattr, eW1, eb1, eW2, eb2,
                                                    aW, ab, pW2, Pa, Pb,
                                                    agg, aggT, cnt, mOut, nEdges);
    node_out_kernel<<<nNodes / 16, 256, 0, stream>>>(h, pos, nW1, nb1, nW2, nb2,
                                                     agg, aggT, cnt, hOut, posOut);
}